// CropToBBox_59992103190909
// MI455X (gfx1250) — compile-verified
//
#include <hip/hip_runtime.h>

// CropToBBox / crop_and_resize for MI455X (gfx1250).
//
// Memory-bound gather (≈240 MB traffic -> ~10 us at 23.3 TB/s); no matrix
// structure, so WMMA is intentionally not used (feeding the matrix pipes would
// add weight-matrix traffic and multiply zeros). Optimizations: wave32-sized
// blocks (8 waves/block), block-uniform box math on the scalar unit with a
// single unconditional s_load_b128 of the bbox row (no scalar branches),
// 12B-contiguous corner gathers (b96-class vmem), coalesced output stores,
// and gfx1250 global_prefetch_b8 for the next output row's input row.

#define OUT_H 224
#define OUT_W 224
#define IMG_H 512
#define IMG_W 512
#define NCHAN 3
#define FACTOR 1.2f
#define THREADS 256
#define BLOCKS_PER_IMAGE ((OUT_H * OUT_W) / THREADS) // 50176/256 = 196 exactly

__device__ __forceinline__ float clip01(float v) {
    return fminf(fmaxf(v, 0.0f), 1.0f);
}

__global__ __launch_bounds__(THREADS) void crop_resize_kernel(
    const float* __restrict__ threshold,  // (N,1)
    const float* __restrict__ bboxes,     // (N,4), rows 16B-aligned
    const float* __restrict__ images,     // (N,512,512,3)
    float* __restrict__ out)              // (N,224,224,3)
{
    // Block-uniform image index: bbox/threshold reads become scalar loads and
    // the whole box computation runs once on the SALU/uniform path.
    const int n   = blockIdx.x / BLOCKS_PER_IMAGE;
    const int rem = blockIdx.x - n * BLOCKS_PER_IMAGE;
    const int pix = rem * THREADS + threadIdx.x;     // 0..50175 within image
    const int i   = pix / OUT_W;                     // output row
    const int j   = pix - i * OUT_W;                 // output col

    // ---- process_bbox (uniform, straight-line) ----
    // Unconditional 16B row load -> single s_load_b128, then cselects.
    const float4 bb = *reinterpret_cast<const float4*>(bboxes + (size_t)n * 4);
    const float th  = threshold[n];
    const bool  def = th < 0.5f;
    const float f0  = def ? 0.0f : bb.x; // x1
    const float f1  = def ? 1.0f : bb.y; // y1
    const float f2  = def ? 0.0f : bb.z; // x2
    const float f3  = def ? 1.0f : bb.w; // y2

    // _resize_side on (x1,x2)
    const float nsx = (f2 - f0) * FACTOR;
    const float cx  = (f0 + f2) * 0.5f;
    const float nx1 = clip01(cx - nsx * 0.5f);
    const float nx2 = clip01(cx + nsx * 0.5f);
    // _resize_side on (y1,y2)
    const float nsy = (f3 - f1) * FACTOR;
    const float cy  = (f1 + f3) * 0.5f;
    const float ny1 = clip01(cy - nsy * 0.5f);
    const float ny2 = clip01(cy + nsy * 0.5f);

    // Reference quirk: crop_and_resize unpacks boxes as [y1,x1,y2,x2] but
    // process_bbox stacked [nx1,ny1,nx2,ny2]. So y-sampling uses the nx pair.
    const float by1 = nx1, by2 = nx2;   // vertical sampling bounds
    const float bx1 = ny1, bx2 = ny2;   // horizontal sampling bounds

    const float ystep = (by2 - by1) * 511.0f / 223.0f;
    const float xstep = (bx2 - bx1) * 511.0f / 223.0f;
    const float ys = by1 * 511.0f + (float)i * ystep;
    const float xs = bx1 * 511.0f + (float)j * xstep;

    const bool valid = (ys >= 0.0f) && (ys <= 511.0f) &&
                       (xs >= 0.0f) && (xs <= 511.0f);

    const float tyf = floorf(ys);
    const float byf = ceilf(ys);
    const float yl  = ys - tyf;
    const float txf = floorf(xs);
    const float bxf = ceilf(xs);
    const float xl  = xs - txf;

    const int tyi = (int)fminf(fmaxf(tyf, 0.0f), 511.0f);
    const int byi = (int)fminf(fmaxf(byf, 0.0f), 511.0f);
    const int txi = (int)fminf(fmaxf(txf, 0.0f), 511.0f);
    const int bxi = (int)fminf(fmaxf(bxf, 0.0f), 511.0f);

    const float* __restrict__ img  = images + (size_t)n * (IMG_H * IMG_W * NCHAN);
    const float* __restrict__ rowT = img + (size_t)tyi * (IMG_W * NCHAN);
    const float* __restrict__ rowB = img + (size_t)byi * (IMG_W * NCHAN);

    // gfx1250 prefetch: pull the next output row's top input row toward the
    // caches (global_prefetch_b8; speculative, address always in-range).
    {
        const float ysn = ys + ystep;
        const int   tyn = (int)fminf(fmaxf(floorf(ysn), 0.0f), 511.0f);
        __builtin_prefetch(img + (size_t)tyn * (IMG_W * NCHAN) + txi * NCHAN, 0, 3);
    }

    // 4 corner gathers, 3 contiguous floats each (12 B -> wide vmem loads).
    float o[NCHAN];
#pragma unroll
    for (int c = 0; c < NCHAN; ++c) {
        const float tl = rowT[txi * NCHAN + c];
        const float tr = rowT[bxi * NCHAN + c];
        const float bl = rowB[txi * NCHAN + c];
        const float br = rowB[bxi * NCHAN + c];
        const float top = tl + (tr - tl) * xl;
        const float bot = bl + (br - bl) * xl;
        o[c] = top + (bot - top) * yl;
    }

    float* __restrict__ op = out + ((size_t)n * (OUT_H * OUT_W) + (size_t)pix) * NCHAN;
#pragma unroll
    for (int c = 0; c < NCHAN; ++c) {
        op[c] = valid ? o[c] : 0.0f;
    }
}

extern "C" void kernel_launch(void* const* d_in, const int* in_sizes, int n_in,
                              void* d_out, int out_size, void* d_ws, size_t ws_size,
                              hipStream_t stream) {
    (void)n_in; (void)d_ws; (void)ws_size; (void)out_size;
    const float* threshold = (const float*)d_in[0]; // (N,1)
    const float* bboxes    = (const float*)d_in[1]; // (N,4)
    const float* images    = (const float*)d_in[2]; // (N,512,512,3)
    float*       out       = (float*)d_out;         // (N,224,224,3)

    const int N = in_sizes[0]; // threshold has N elements (N,1)
    const dim3 grid(N * BLOCKS_PER_IMAGE);
    const dim3 block(THREADS);
    crop_resize_kernel<<<grid, block, 0, stream>>>(threshold, bboxes, images, out);
}